// GConv_81372450390360
// MI455X (gfx1250) — compile-verified
//
#include <hip/hip_runtime.h>
#include <hip/hip_bf16.h>
#include <math.h>

#define N_NODES 50000
#define N_EDGES 800000
#define D 128
#define OUTC 384
#define EPS_GN 1e-5f

typedef __attribute__((ext_vector_type(2))) float v2f;
typedef __attribute__((ext_vector_type(8))) float v8f;

// Hardware GLOBAL_ATOMIC_ADD_F32 (CDNA5 has native f32 fadd atomics; coarse-grained
// device memory only, which is what hipMalloc'd d_ws is).
__device__ __forceinline__ void atomAddF(float* p, float v) {
  unsafeAtomicAdd(p, v);
}

// ---------------- degree / edge norm ----------------
__global__ void k_deg_init(float* deg, int n) {
  int i = blockIdx.x * blockDim.x + threadIdx.x;
  if (i < n) deg[i] = 1.0f;  // self loop contributes 1
}
__global__ void k_deg_count(const int* __restrict__ col, float* deg, int e) {
  int i = blockIdx.x * blockDim.x + threadIdx.x;
  if (i < e) atomAddF(&deg[col[i]], 1.0f);
}
__global__ void k_deg_rsqrt(float* deg, int n) {
  int i = blockIdx.x * blockDim.x + threadIdx.x;
  if (i < n) deg[i] = rsqrtf(deg[i]);   // deg >= 1 always
}
__global__ void k_edge_norm(const int* __restrict__ row, const int* __restrict__ col,
                            const float* __restrict__ dinv, float* __restrict__ nrm, int e) {
  int i = blockIdx.x * blockDim.x + threadIdx.x;
  if (i < e) nrm[i] = dinv[row[i]] * dinv[col[i]];
}

// ---------------- WMMA f32 GEMM: H[N,128] = X[N,128] @ W[128,128] ----------------
// One wave computes one 16x16 tile. K stepped by 4 with V_WMMA_F32_16X16X4_F32.
__global__ __launch_bounds__(32) void k_gemm(const float* __restrict__ X,
                                             const float* __restrict__ W,
                                             float* __restrict__ H) {
  const int lane = threadIdx.x;
  const int m    = lane & 15;            // M index (A) / N index (B,D)
  const int kb   = (lane >> 4) << 1;     // lanes 0-15 -> k{0,1}, lanes 16-31 -> k{2,3}
  const int rowBase = blockIdx.x * 16;
  const int colBase = blockIdx.y * 16;
  const float* xrow = X + (size_t)(rowBase + m) * D;

  v8f acc = {};
  #pragma unroll 4
  for (int k0 = 0; k0 < D; k0 += 4) {
    v2f a = *reinterpret_cast<const v2f*>(xrow + k0 + kb);        // A: 16x4 fragment
    v2f b;
    b.x = W[(size_t)(k0 + kb)     * D + colBase + m];             // B: 4x16 fragment
    b.y = W[(size_t)(k0 + kb + 1) * D + colBase + m];
    acc = __builtin_amdgcn_wmma_f32_16x16x4_f32(false, a, false, b,
                                                (short)0, acc, false, false);
  }
  // D layout: VGPR r holds M = r + 8*(lane>=16), N = lane&15
  const int outCol = colBase + m;
  const int rOff   = (lane >> 4) * 8;
  #pragma unroll
  for (int r = 0; r < 8; ++r)
    H[(size_t)(rowBase + rOff + r) * D + outCol] = acc[r];
}

// ---------------- aggregation ----------------
// float4-vectorized: one thread = one node x 4-channel chunk (n * 32 threads)
__global__ void k_agg_init(const float* __restrict__ H, const float* __restrict__ dinv,
                           const float* __restrict__ bias, float* __restrict__ AGG, int n) {
  int idx = blockIdx.x * blockDim.x + threadIdx.x;
  if (idx >= n * (D / 4)) return;
  int i  = idx >> 5;
  int c0 = (idx & 31) << 2;
  float di = dinv[i];
  float s  = di * di;
  float4 h = *reinterpret_cast<const float4*>(H + (size_t)idx * 4);
  float4 b = *reinterpret_cast<const float4*>(bias + c0);
  float4 r;
  r.x = s * h.x + b.x; r.y = s * h.y + b.y;
  r.z = s * h.z + b.z; r.w = s * h.w + b.w;
  *reinterpret_cast<float4*>(AGG + (size_t)idx * 4) = r;
}

__global__ void k_scatter(const int* __restrict__ row, const int* __restrict__ col,
                          const float* __restrict__ nrm, const float* __restrict__ H,
                          float* __restrict__ AGG, int e) {
  unsigned idx = blockIdx.x * blockDim.x + threadIdx.x;  // e*32 work items
  unsigned ed  = idx >> 5;
  if (ed >= (unsigned)e) return;
  int c0 = (idx & 31) << 2;   // 4-channel chunk
  int r = row[ed], c = col[ed];
  float w = nrm[ed];
  const float4 h = *reinterpret_cast<const float4*>(H + (size_t)r * D + c0);
  float* dst = AGG + (size_t)c * D + c0;
  atomAddF(dst + 0, w * h.x);
  atomAddF(dst + 1, w * h.y);
  atomAddF(dst + 2, w * h.z);
  atomAddF(dst + 3, w * h.w);
}

// ---------------- GraphNorm ----------------
__global__ void k_zero(float* p, int n) {
  int i = blockIdx.x * blockDim.x + threadIdx.x;
  if (i < n) p[i] = 0.0f;
}

// Per-channel column reduction; centered=1 accumulates (x - a*mean)^2.
__global__ void k_col_reduce(const float* __restrict__ X, const float* __restrict__ mean,
                             const float* __restrict__ alpha, float* __restrict__ sums,
                             int n, int centered) {
  __shared__ float4 sm[256];
  int cg = threadIdx.x & 31;   // channel group of 4
  int ry = threadIdx.x >> 5;   // 0..7
  int c0 = cg << 2;
  float4 off = make_float4(0.f, 0.f, 0.f, 0.f);
  if (centered) {
    off.x = alpha[c0 + 0] * mean[c0 + 0];
    off.y = alpha[c0 + 1] * mean[c0 + 1];
    off.z = alpha[c0 + 2] * mean[c0 + 2];
    off.w = alpha[c0 + 3] * mean[c0 + 3];
  }
  float4 acc = make_float4(0.f, 0.f, 0.f, 0.f);
  int r0 = blockIdx.x * 256;
  int rend = min(r0 + 256, n);
  for (int r = r0 + ry; r < rend; r += 8) {
    float4 v = *reinterpret_cast<const float4*>(X + (size_t)r * D + c0);
    if (centered) {
      v.x -= off.x; v.y -= off.y; v.z -= off.z; v.w -= off.w;
      acc.x += v.x * v.x; acc.y += v.y * v.y; acc.z += v.z * v.z; acc.w += v.w * v.w;
    } else {
      acc.x += v.x; acc.y += v.y; acc.z += v.z; acc.w += v.w;
    }
  }
  sm[threadIdx.x] = acc;
  __syncthreads();
  if (ry == 0) {
    float4 t = sm[cg];
    #pragma unroll
    for (int k = 1; k < 8; ++k) {
      float4 u = sm[k * 32 + cg];
      t.x += u.x; t.y += u.y; t.z += u.z; t.w += u.w;
    }
    atomAddF(&sums[c0 + 0], t.x);
    atomAddF(&sums[c0 + 1], t.y);
    atomAddF(&sums[c0 + 2], t.z);
    atomAddF(&sums[c0 + 3], t.w);
  }
}

__global__ void k_finalize_mean(const float* __restrict__ sums, float* mean, float invn) {
  int c = threadIdx.x;
  mean[c] = sums[c] * invn;
}
__global__ void k_finalize_rsig(const float* __restrict__ sums, float* rsig, float invn) {
  int c = threadIdx.x;
  rsig[c] = rsqrtf(sums[c] * invn + EPS_GN);
}

// normalize, write concat slice (pre-ELU), and write ELU(act) for next layer.
// float4-vectorized: one thread = one node x 4-channel chunk.
__global__ void k_gnorm_apply(const float* __restrict__ AGG, const float* __restrict__ mean,
                              const float* __restrict__ rsig,
                              const float* __restrict__ gw, const float* __restrict__ gb,
                              const float* __restrict__ ga,
                              float* __restrict__ out, int colOff,
                              float* __restrict__ act, int n) {
  int idx = blockIdx.x * blockDim.x + threadIdx.x;
  if (idx >= n * (D / 4)) return;
  int i  = idx >> 5;
  int c0 = (idx & 31) << 2;
  float4 v  = *reinterpret_cast<const float4*>(AGG  + (size_t)idx * 4);
  float4 m4 = *reinterpret_cast<const float4*>(mean + c0);
  float4 s4 = *reinterpret_cast<const float4*>(rsig + c0);
  float4 w4 = *reinterpret_cast<const float4*>(gw   + c0);
  float4 b4 = *reinterpret_cast<const float4*>(gb   + c0);
  float4 a4 = *reinterpret_cast<const float4*>(ga   + c0);
  float4 y;
  y.x = w4.x * (v.x - a4.x * m4.x) * s4.x + b4.x;
  y.y = w4.y * (v.y - a4.y * m4.y) * s4.y + b4.y;
  y.z = w4.z * (v.z - a4.z * m4.z) * s4.z + b4.z;
  y.w = w4.w * (v.w - a4.w * m4.w) * s4.w + b4.w;
  *reinterpret_cast<float4*>(out + (size_t)i * OUTC + colOff + c0) = y;
  float4 ev;
  ev.x = (y.x > 0.f) ? y.x : (__expf(y.x) - 1.0f);   // ELU alpha=1
  ev.y = (y.y > 0.f) ? y.y : (__expf(y.y) - 1.0f);
  ev.z = (y.z > 0.f) ? y.z : (__expf(y.z) - 1.0f);
  ev.w = (y.w > 0.f) ? y.w : (__expf(y.w) - 1.0f);
  *reinterpret_cast<float4*>(act + (size_t)idx * 4) = ev;
}

__global__ void k_copy_out3(const float* __restrict__ AGG, float* __restrict__ out, int n) {
  int idx = blockIdx.x * blockDim.x + threadIdx.x;
  if (idx >= n * (D / 4)) return;
  int i  = idx >> 5;
  int c0 = (idx & 31) << 2;
  float4 v = *reinterpret_cast<const float4*>(AGG + (size_t)idx * 4);
  *reinterpret_cast<float4*>(out + (size_t)i * OUTC + 256 + c0) = v;
}

extern "C" void kernel_launch(void* const* d_in, const int* in_sizes, int n_in,
                              void* d_out, int out_size, void* d_ws, size_t ws_size,
                              hipStream_t stream) {
  const float* x    = (const float*)d_in[0];
  const int*   ei   = (const int*)d_in[1];
  const float* W1   = (const float*)d_in[2];
  const float* b1   = (const float*)d_in[3];
  const float* W2   = (const float*)d_in[4];
  const float* b2   = (const float*)d_in[5];
  const float* W3   = (const float*)d_in[6];
  const float* b3   = (const float*)d_in[7];
  const float* gn1w = (const float*)d_in[8];
  const float* gn1b = (const float*)d_in[9];
  const float* gn1a = (const float*)d_in[10];
  const float* gn2w = (const float*)d_in[11];
  const float* gn2b = (const float*)d_in[12];
  const float* gn2a = (const float*)d_in[13];

  const int n = N_NODES, e = N_EDGES;
  const int* row = ei;        // edge_index[0] = sources
  const int* col = ei + e;    // edge_index[1] = targets

  float* ws   = (float*)d_ws;
  float* dinv = ws;  ws += n;
  float* nrm  = ws;  ws += e;
  float* H    = ws;  ws += (size_t)n * D;
  float* AGG  = ws;  ws += (size_t)n * D;
  float* ACT  = ws;  ws += (size_t)n * D;
  float* sums = ws;  ws += D;
  float* mean = ws;  ws += D;
  float* rsig = ws;  ws += D;

  float* out = (float*)d_out;

  const dim3 b256(256);
  const int gN    = (n + 255) / 256;
  const int gE    = (e + 255) / 256;
  const int gND4  = (n * (D / 4) + 255) / 256;   // float4 elementwise kernels
  const int gScat = (e * 32 + 255) / 256;
  const dim3 gGemm(n / 16, D / 16);
  const int gRed  = (n + 255) / 256;
  const float invn = 1.0f / (float)n;

  // degree + symmetric normalization
  k_deg_init <<<gN, b256, 0, stream>>>(dinv, n);
  k_deg_count<<<gE, b256, 0, stream>>>(col, dinv, e);
  k_deg_rsqrt<<<gN, b256, 0, stream>>>(dinv, n);
  k_edge_norm<<<gE, b256, 0, stream>>>(row, col, dinv, nrm, e);

  // ---- layer 1 ----
  k_gemm    <<<gGemm, 32, 0, stream>>>(x, W1, H);
  k_agg_init<<<gND4, b256, 0, stream>>>(H, dinv, b1, AGG, n);
  k_scatter <<<gScat, b256, 0, stream>>>(row, col, nrm, H, AGG, e);
  k_zero    <<<1, 128, 0, stream>>>(sums, D);
  k_col_reduce<<<gRed, b256, 0, stream>>>(AGG, nullptr, nullptr, sums, n, 0);
  k_finalize_mean<<<1, 128, 0, stream>>>(sums, mean, invn);
  k_zero    <<<1, 128, 0, stream>>>(sums, D);
  k_col_reduce<<<gRed, b256, 0, stream>>>(AGG, mean, gn1a, sums, n, 1);
  k_finalize_rsig<<<1, 128, 0, stream>>>(sums, rsig, invn);
  k_gnorm_apply<<<gND4, b256, 0, stream>>>(AGG, mean, rsig, gn1w, gn1b, gn1a, out, 0, ACT, n);

  // ---- layer 2 ----
  k_gemm    <<<gGemm, 32, 0, stream>>>(ACT, W2, H);
  k_agg_init<<<gND4, b256, 0, stream>>>(H, dinv, b2, AGG, n);
  k_scatter <<<gScat, b256, 0, stream>>>(row, col, nrm, H, AGG, e);
  k_zero    <<<1, 128, 0, stream>>>(sums, D);
  k_col_reduce<<<gRed, b256, 0, stream>>>(AGG, nullptr, nullptr, sums, n, 0);
  k_finalize_mean<<<1, 128, 0, stream>>>(sums, mean, invn);
  k_zero    <<<1, 128, 0, stream>>>(sums, D);
  k_col_reduce<<<gRed, b256, 0, stream>>>(AGG, mean, gn2a, sums, n, 1);
  k_finalize_rsig<<<1, 128, 0, stream>>>(sums, rsig, invn);
  k_gnorm_apply<<<gND4, b256, 0, stream>>>(AGG, mean, rsig, gn2w, gn2b, gn2a, out, 128, ACT, n);

  // ---- layer 3 (no norm, no activation) ----
  k_gemm    <<<gGemm, 32, 0, stream>>>(ACT, W3, H);
  k_agg_init<<<gND4, b256, 0, stream>>>(H, dinv, b3, AGG, n);
  k_scatter <<<gScat, b256, 0, stream>>>(row, col, nrm, H, AGG, e);
  k_copy_out3<<<gND4, b256, 0, stream>>>(AGG, out, n);
}